// PNA4_9294309228816
// MI455X (gfx1250) — compile-verified
//
#include <hip/hip_runtime.h>
#include <hip/hip_bf16.h>

// ---------------------------------------------------------------------------
// PNA (3x PNAConv + mean-pool + MLP head) for MI455X / gfx1250, wave32.
// All GEMMs use V_WMMA_F32_16X16X4_F32 (full f32 == reference precision;
// the workload is L2/atomic bound, not FLOP bound, so f32 matrix path is the
// right choice). Edge messages are fused with scatter aggregation via
// global f32-add atomics and monotone-uint-mapped u32 min/max atomics.
// ---------------------------------------------------------------------------

typedef __attribute__((ext_vector_type(2))) float v2f;
typedef __attribute__((ext_vector_type(8))) float v8f;

#if defined(__HIP_DEVICE_COMPILE__)
#if !__has_builtin(__builtin_amdgcn_wmma_f32_16x16x4_f32)
#error "gfx1250 f32 16x16x4 WMMA builtin not available on this toolchain"
#endif
#endif

__device__ __forceinline__ v8f wmma4(v2f a, v2f b, v8f c) {
  // D = A(16x4 f32) * B(4x16 f32) + C(16x16 f32)
  return __builtin_amdgcn_wmma_f32_16x16x4_f32(false, a, false, b, (short)0, c,
                                               false, false);
}

// monotone f32 <-> u32 mapping so u32 atomic min/max implement float min/max
__device__ __forceinline__ unsigned f2mono(float f) {
  unsigned u = __float_as_uint(f);
  return (u & 0x80000000u) ? ~u : (u | 0x80000000u);
}
__device__ __forceinline__ float mono2f(unsigned u) {
  return __uint_as_float((u & 0x80000000u) ? (u ^ 0x80000000u) : ~u);
}

// ---------------------------------------------------------------------------
// small prep kernels
// ---------------------------------------------------------------------------

// pad x [N,7] -> xpad [N,16] (zeros in cols 7..15)
__global__ void k_pad_x(const float* __restrict__ x, float* __restrict__ xp,
                        int Nn) {
  int i = blockIdx.x * blockDim.x + threadIdx.x;
  if (i >= Nn * 16) return;
  int node = i >> 4, c = i & 15;
  xp[i] = (c < 7) ? x[node * 7 + c] : 0.0f;
}

// reset aggregation buffers (sum, sumsq, mapped-min, mapped-max, count)
__global__ void k_init_agg(float* __restrict__ s, float* __restrict__ s2,
                           unsigned* __restrict__ mn, unsigned* __restrict__ mx,
                           unsigned* __restrict__ cnt, int Nn) {
  int i = blockIdx.x * blockDim.x + threadIdx.x;
  if (i < Nn * 64) {
    s[i] = 0.0f; s2[i] = 0.0f;
    mn[i] = 0xFFFFFFFFu;   // mapped +inf
    mx[i] = 0u;            // mapped -inf
  }
  if (i < Nn) cnt[i] = 0u;
}

// layer-1 repack: Wcat1[32x16] = padded [Wd;Ws] of Wpre1(21x7),
// ve1/ve0 = folded edge-encoder, Wpp[208x64] = padded/reordered Wpost1(91x64)
__global__ void k_prep_l1(const float* __restrict__ Wpre,
                          const float* __restrict__ We,
                          const float* __restrict__ be,
                          const float* __restrict__ bpre,
                          const float* __restrict__ Wpost,
                          float* __restrict__ Wcat, float* __restrict__ ve0,
                          float* __restrict__ ve1, float* __restrict__ Wpp) {
  int tid = threadIdx.x;
  for (int i = tid; i < 32 * 16; i += 256) Wcat[i] = 0.0f;
  for (int i = tid; i < 208 * 64; i += 256) Wpp[i] = 0.0f;
  if (tid < 16) {
    float a1 = 0.0f, a0 = 0.0f;
    if (tid < 7) {
      for (int k = 0; k < 7; ++k) {
        float w = Wpre[(14 + k) * 7 + tid];  // edge-encoder rows of Wpre
        a1 += We[k] * w;
        a0 += be[k] * w;
      }
      a0 += bpre[tid];
    }
    ve1[tid] = a1;
    ve0[tid] = a0;
  }
  __syncthreads();
  for (int i = tid; i < 49; i += 256) {
    int r = i / 7, c = i % 7;
    Wcat[r * 16 + c]        = Wpre[r * 7 + c];        // x[dst] rows
    Wcat[(16 + r) * 16 + c] = Wpre[(7 + r) * 7 + c];  // x[src] rows
  }
  for (int i = tid; i < 91 * 64; i += 256) {
    int r = i >> 6, j = i & 63;
    int pr;
    if (r < 7)        pr = r;                                        // x
    else if (r < 35) { int q = r - 7;  pr = 16  + (q / 7) * 16 + q % 7; }
    else if (r < 63) { int q = r - 35; pr = 80  + (q / 7) * 16 + q % 7; }
    else             { int q = r - 63; pr = 144 + (q / 7) * 16 + q % 7; }
    Wpp[pr * 64 + j] = Wpost[i];
  }
}

// layers 2/3: fold edge encoder through Wpre rows [128,192)
__global__ void k_prep_ve(const float* __restrict__ Wpre,
                          const float* __restrict__ We,
                          const float* __restrict__ be,
                          const float* __restrict__ bpre,
                          float* __restrict__ ve0, float* __restrict__ ve1) {
  int c = threadIdx.x;
  if (c >= 64) return;
  float a1 = 0.0f, a0 = 0.0f;
  for (int k = 0; k < 64; ++k) {
    float w = Wpre[(128 + k) * 64 + c];
    a1 += We[k] * w;
    a0 += be[k] * w;
  }
  ve1[c] = a1;
  ve0[c] = a0 + bpre[c];
}

// ---------------------------------------------------------------------------
// edge kernel: per 16-edge tile, gather [x[dst]|x[src]] (16 x 2F) into LDS,
// WMMA-GEMM against W (2F x F), add rank-1 edge term, scatter-aggregate with
// atomics. One tile per wave, 4 waves per block.
// ---------------------------------------------------------------------------
template <int F>
__global__ __launch_bounds__(128) void k_edge(
    const float* __restrict__ xin, const int* __restrict__ eidx,
    const float* __restrict__ eattr, const float* __restrict__ W,
    const float* __restrict__ ve0, const float* __restrict__ ve1,
    float* __restrict__ gS, float* __restrict__ gS2,
    unsigned* __restrict__ gMn, unsigned* __restrict__ gMx,
    unsigned* __restrict__ gCnt, int Ee, int etiles) {
  constexpr int K = 2 * F;
  constexpr int ASTR = K + 4;     // pad: conflict-free column reads, 16B-aligned rows
  constexpr int NT = F / 16;
  constexpr int SEGS = K / 4;     // float4 segments per A row
  __shared__ float sA[4][16 * ASTR];
  __shared__ int sIdx[4][32];     // [0..15]=dst, [16..31]=src
  __shared__ float sAttr[4][16];

  const int w = threadIdx.x >> 5, lane = threadIdx.x & 31;
  const int tile = blockIdx.x * 4 + w;
  if (tile >= etiles) return;     // wave-uniform
  const long e0 = (long)tile * 16;

  if (lane < 16) {
    long e = e0 + lane;
    bool v = e < Ee;
    sIdx[w][lane] = v ? eidx[(long)Ee + e] : 0;  // edge_index[1] = dst
    sAttr[w][lane] = v ? eattr[e] : 0.0f;
  } else {
    long e = e0 + (lane - 16);
    sIdx[w][lane] = (e < Ee) ? eidx[e] : 0;      // edge_index[0] = src
  }
  __builtin_amdgcn_wave_barrier();

  // cooperative gather: row m = [x[dst[m]] (F) | x[src[m]] (F)]
  for (int i = lane; i < 16 * SEGS; i += 32) {
    int m = i / SEGS, seg = i % SEGS;
    int node = (seg < SEGS / 2) ? sIdx[w][m] : sIdx[w][16 + m];
    int col4 = (seg < SEGS / 2) ? seg : (seg - SEGS / 2);
    const float4 vv = *(const float4*)(xin + (long)node * F + col4 * 4);
    *(float4*)(&sA[w][m * ASTR + seg * 4]) = vv;
  }
  __builtin_amdgcn_wave_barrier();

  const int n = lane & 15, hi = lane >> 4, m = lane & 15;
  for (int t = 0; t < NT; ++t) {
    v8f acc = {};
#pragma unroll 8
    for (int k0 = 0; k0 < K; k0 += 4) {
      int kk = k0 + 2 * hi;
      v2f a;
      a.x = sA[w][m * ASTR + kk];
      a.y = sA[w][m * ASTR + kk + 1];
      v2f b;
      b.x = W[kk * F + t * 16 + n];
      b.y = W[(kk + 1) * F + t * 16 + n];
      acc = wmma4(a, b, acc);
    }
    float e1 = ve1[t * 16 + n], ec = ve0[t * 16 + n];
#pragma unroll
    for (int v = 0; v < 8; ++v) {
      int row = v + 8 * hi;
      if (e0 + row < Ee) {
        int node = sIdx[w][row];  // dst
        float val = acc[v] + sAttr[w][row] * e1 + ec;
        long ad = (long)node * F + t * 16 + n;
        atomicAdd(&gS[ad], val);
        atomicAdd(&gS2[ad], val * val);
        unsigned u = f2mono(val);
        atomicMin(&gMn[ad], u);
        atomicMax(&gMx[ad], u);
      }
    }
  }
  if (lane < 16 && (e0 + lane) < Ee) atomicAdd(&gCnt[sIdx[w][lane]], 1u);
}

// ---------------------------------------------------------------------------
// node kernel: per 16-node tile, build compact agg tile [16 x 4F] + per-node
// scalers in LDS, WMMA-GEMM [16 x 13F] @ Wpost (scaler copies decoded on the
// fly), + bpost, then [16 x 64] @ Wlin + blin, optional leaky-relu.
// One tile per wave, 2 waves per block.
// ---------------------------------------------------------------------------
template <int F, bool LEAKY>
__global__ __launch_bounds__(64) void k_node(
    const float* __restrict__ xin, const float* __restrict__ gS,
    const float* __restrict__ gS2, const unsigned* __restrict__ gMn,
    const unsigned* __restrict__ gMx, const unsigned* __restrict__ gCnt,
    const float* __restrict__ Wp, const float* __restrict__ bpost,
    const float* __restrict__ Wlin, const float* __restrict__ blin,
    float* __restrict__ xout, int Nn, int ntiles) {
  constexpr int AG = 4 * F;
  constexpr int AGSTR = AG + 4;
  constexpr int A2STR = 68;
  __shared__ float sAgg[2][16 * AGSTR];
  __shared__ float sScl[2][16 * 4];
  __shared__ float sA2[2][16 * A2STR];

  const int w = threadIdx.x >> 5, lane = threadIdx.x & 31;
  const int tile = blockIdx.x * 2 + w;
  if (tile >= ntiles) return;  // wave-uniform
  const int base = tile * 16;

  // phase 1: per-node stats -> LDS (mean | min | max | std), q = a*F + c
  constexpr int PER = (16 * F) / 32;
  for (int i = 0; i < PER; ++i) {
    int idx = i * 32 + lane;
    int r = idx / F, c = idx % F;
    int node = base + r; if (node >= Nn) node = Nn - 1;
    unsigned cn = gCnt[node];
    float denom = fmaxf((float)cn, 1.0f);
    long ad = (long)node * F + c;
    float mean = gS[ad] / denom;
    float msq = gS2[ad] / denom;
    float sd = sqrtf(fmaxf(msq - mean * mean, 0.0f) + 1e-5f);
    float vmn = (cn > 0u) ? mono2f(gMn[ad]) : 0.0f;
    float vmx = (cn > 0u) ? mono2f(gMx[ad]) : 0.0f;
    float* rowp = &sAgg[w][r * AGSTR];
    rowp[c] = mean;
    rowp[F + c] = vmn;
    rowp[2 * F + c] = vmx;
    rowp[3 * F + c] = sd;
  }
  if (lane < 16) {
    int node = base + lane; if (node >= Nn) node = Nn - 1;
    float denom = fmaxf((float)gCnt[node], 1.0f);
    float amp = logf(denom + 1.0f) * 0.35295573f;  // 1/ln(17)
    sScl[w][lane * 4 + 0] = 1.0f;
    sScl[w][lane * 4 + 1] = amp;
    sScl[w][lane * 4 + 2] = 1.0f / amp;
  }
  __builtin_amdgcn_wave_barrier();

  const int n = lane & 15, hi = lane >> 4, m = lane & 15;
  int nodem = base + m; if (nodem >= Nn) nodem = Nn - 1;

  // GEMM1: [16 x 13F] @ Wp[13F x 64] + bpost   (A cols: x | agg | agg*amp | agg/amp)
  for (int t = 0; t < 4; ++t) {
    v8f acc = {};
#pragma unroll
    for (int k0 = 0; k0 < F; k0 += 4) {  // x region
      int kk = k0 + 2 * hi;
      v2f a;
      a.x = xin[(long)nodem * F + kk];
      a.y = xin[(long)nodem * F + kk + 1];
      v2f b;
      b.x = Wp[kk * 64 + t * 16 + n];
      b.y = Wp[(kk + 1) * 64 + t * 16 + n];
      acc = wmma4(a, b, acc);
    }
#pragma unroll
    for (int s = 0; s < 3; ++s) {  // identity / amplification / attenuation
      float scl = sScl[w][m * 4 + s];
#pragma unroll 4
      for (int k0 = 0; k0 < AG; k0 += 4) {
        int kk = k0 + 2 * hi;
        int j = F + s * AG + kk;
        v2f a;
        a.x = sAgg[w][m * AGSTR + kk] * scl;
        a.y = sAgg[w][m * AGSTR + kk + 1] * scl;
        v2f b;
        b.x = Wp[j * 64 + t * 16 + n];
        b.y = Wp[(j + 1) * 64 + t * 16 + n];
        acc = wmma4(a, b, acc);
      }
    }
    float bp = bpost[t * 16 + n];
#pragma unroll
    for (int vv = 0; vv < 8; ++vv)
      sA2[w][(vv + 8 * hi) * A2STR + t * 16 + n] = acc[vv] + bp;
  }
  __builtin_amdgcn_wave_barrier();

  // GEMM2: [16 x 64] @ Wlin[64 x 64] + blin (+ leaky)
  for (int t = 0; t < 4; ++t) {
    v8f acc = {};
#pragma unroll 8
    for (int k0 = 0; k0 < 64; k0 += 4) {
      int kk = k0 + 2 * hi;
      v2f a;
      a.x = sA2[w][m * A2STR + kk];
      a.y = sA2[w][m * A2STR + kk + 1];
      v2f b;
      b.x = Wlin[kk * 64 + t * 16 + n];
      b.y = Wlin[(kk + 1) * 64 + t * 16 + n];
      acc = wmma4(a, b, acc);
    }
    float bl = blin[t * 16 + n];
#pragma unroll
    for (int vv = 0; vv < 8; ++vv) {
      int node = base + vv + 8 * hi;
      if (node < Nn) {
        float val = acc[vv] + bl;
        if (LEAKY) val = (val > 0.0f) ? val : 0.01f * val;
        xout[(long)node * 64 + t * 16 + n] = val;
      }
    }
  }
}

// ---------------------------------------------------------------------------
// pooling + head
// ---------------------------------------------------------------------------
__global__ void k_pool_zero(float* __restrict__ gsum, float* __restrict__ gcnt) {
  int tid = threadIdx.x;
  if (tid < 1024) gsum[tid] = 0.0f;
  if (tid < 16) gcnt[tid] = 0.0f;
}

// batch is sorted: accumulate runs locally, flush on graph change
__global__ __launch_bounds__(256) void k_pool(const float* __restrict__ h,
                                              const int* __restrict__ batch,
                                              float* __restrict__ gsum,
                                              float* __restrict__ gcnt, int Nn) {
  int ch = threadIdx.x & 63;
  int grp = threadIdx.x >> 6;
  int start = blockIdx.x * 1024 + grp * 256;
  float acc = 0.0f, ca = 0.0f;
  int curg = -1;
  for (int i = 0; i < 256; ++i) {
    int node = start + i;
    if (node >= Nn) break;
    int g = batch[node];
    if (g != curg) {
      if (curg >= 0) {
        atomicAdd(&gsum[curg * 64 + ch], acc);
        if (ch == 0) atomicAdd(&gcnt[curg], ca);
      }
      curg = g; acc = 0.0f; ca = 0.0f;
    }
    acc += h[(long)node * 64 + ch];
    ca += 1.0f;
  }
  if (curg >= 0) {
    atomicAdd(&gsum[curg * 64 + ch], acc);
    if (ch == 0) atomicAdd(&gcnt[curg], ca);
  }
}

__global__ __launch_bounds__(1024) void k_head(
    const float* __restrict__ gsum, const float* __restrict__ gcnt,
    const float* __restrict__ Wh1, const float* __restrict__ bh1,
    const float* __restrict__ Wh2, const float* __restrict__ bh2,
    float* __restrict__ out) {
  __shared__ float g[16 * 64];
  __shared__ float h[16 * 32];
  int tid = threadIdx.x;
  {
    int i = tid >> 6;
    g[tid] = gsum[tid] / fmaxf(gcnt[i], 1.0f);
  }
  __syncthreads();
  if (tid < 512) {
    int i = tid >> 5, j = tid & 31;
    float a = bh1[j];
    for (int k = 0; k < 64; ++k) a += g[i * 64 + k] * Wh1[k * 32 + j];
    h[i * 32 + j] = fmaxf(a, 0.0f);
  }
  __syncthreads();
  if (tid < 80) {
    int i = tid / 5, j = tid % 5;
    float a = bh2[j];
    for (int k = 0; k < 32; ++k) a += h[i * 32 + k] * Wh2[k * 5 + j];
    out[i * 5 + j] = a;
  }
}

// ---------------------------------------------------------------------------
// launcher
// ---------------------------------------------------------------------------
extern "C" void kernel_launch(void* const* d_in, const int* in_sizes, int n_in,
                              void* d_out, int out_size, void* d_ws,
                              size_t ws_size, hipStream_t stream) {
  (void)n_in; (void)out_size; (void)ws_size;
  const float* x      = (const float*)d_in[0];
  const int*   eidx   = (const int*)d_in[1];
  const float* eattr  = (const float*)d_in[2];
  const int*   batch  = (const int*)d_in[3];
  const float* We1    = (const float*)d_in[4];
  const float* be1    = (const float*)d_in[5];
  const float* Wpre1  = (const float*)d_in[6];
  const float* bpre1  = (const float*)d_in[7];
  const float* Wpost1 = (const float*)d_in[8];
  const float* bpost1 = (const float*)d_in[9];
  const float* Wlin1  = (const float*)d_in[10];
  const float* blin1  = (const float*)d_in[11];
  const float* We2    = (const float*)d_in[12];
  const float* be2    = (const float*)d_in[13];
  const float* Wpre2  = (const float*)d_in[14];
  const float* bpre2  = (const float*)d_in[15];
  const float* Wpost2 = (const float*)d_in[16];
  const float* bpost2 = (const float*)d_in[17];
  const float* Wlin2  = (const float*)d_in[18];
  const float* blin2  = (const float*)d_in[19];
  const float* We3    = (const float*)d_in[20];
  const float* be3    = (const float*)d_in[21];
  const float* Wpre3  = (const float*)d_in[22];
  const float* bpre3  = (const float*)d_in[23];
  const float* Wpost3 = (const float*)d_in[24];
  const float* bpost3 = (const float*)d_in[25];
  const float* Wlin3  = (const float*)d_in[26];
  const float* blin3  = (const float*)d_in[27];
  const float* Wh1    = (const float*)d_in[28];
  const float* bh1    = (const float*)d_in[29];
  const float* Wh2    = (const float*)d_in[30];
  const float* bh2    = (const float*)d_in[31];
  float* out = (float*)d_out;

  const int Nn = in_sizes[0] / 7;
  const int Ee = in_sizes[1] / 2;

  // carve workspace (~160 MB total; everything re-initialized every call)
  char* p = (char*)d_ws;
  auto carve = [&](size_t bytes) -> void* {
    void* q = (void*)p;
    p += (bytes + 255) & ~(size_t)255;
    return q;
  };
  float*    xpad  = (float*)carve((size_t)Nn * 16 * sizeof(float));
  float*    hA    = (float*)carve((size_t)Nn * 64 * sizeof(float));
  float*    hB    = (float*)carve((size_t)Nn * 64 * sizeof(float));
  float*    gS    = (float*)carve((size_t)Nn * 64 * sizeof(float));
  float*    gS2   = (float*)carve((size_t)Nn * 64 * sizeof(float));
  unsigned* gMn   = (unsigned*)carve((size_t)Nn * 64 * sizeof(unsigned));
  unsigned* gMx   = (unsigned*)carve((size_t)Nn * 64 * sizeof(unsigned));
  unsigned* gCnt  = (unsigned*)carve((size_t)Nn * sizeof(unsigned));
  float*    Wcat1 = (float*)carve(32 * 16 * sizeof(float));
  float*    ve0_1 = (float*)carve(16 * sizeof(float));
  float*    ve1_1 = (float*)carve(16 * sizeof(float));
  float*    ve0_2 = (float*)carve(64 * sizeof(float));
  float*    ve1_2 = (float*)carve(64 * sizeof(float));
  float*    ve0_3 = (float*)carve(64 * sizeof(float));
  float*    ve1_3 = (float*)carve(64 * sizeof(float));
  float*    Wpp1  = (float*)carve(208 * 64 * sizeof(float));
  float*    gsum  = (float*)carve(16 * 64 * sizeof(float));
  float*    gcnt  = (float*)carve(16 * sizeof(float));

  k_pad_x<<<(Nn * 16 + 255) / 256, 256, 0, stream>>>(x, xpad, Nn);
  k_prep_l1<<<1, 256, 0, stream>>>(Wpre1, We1, be1, bpre1, Wpost1, Wcat1,
                                   ve0_1, ve1_1, Wpp1);
  k_prep_ve<<<1, 64, 0, stream>>>(Wpre2, We2, be2, bpre2, ve0_2, ve1_2);
  k_prep_ve<<<1, 64, 0, stream>>>(Wpre3, We3, be3, bpre3, ve0_3, ve1_3);

  const int etiles  = (Ee + 15) / 16;
  const int eblocks = (etiles + 3) / 4;
  const int ntiles  = (Nn + 15) / 16;
  const int nblocks = (ntiles + 1) / 2;
  const int iblocks = (Nn * 64 + 255) / 256;

  // layer 1 (F padded 7->16)
  k_init_agg<<<iblocks, 256, 0, stream>>>(gS, gS2, gMn, gMx, gCnt, Nn);
  k_edge<16><<<eblocks, 128, 0, stream>>>(xpad, eidx, eattr, Wcat1, ve0_1,
                                          ve1_1, gS, gS2, gMn, gMx, gCnt, Ee,
                                          etiles);
  k_node<16, true><<<nblocks, 64, 0, stream>>>(xpad, gS, gS2, gMn, gMx, gCnt,
                                               Wpp1, bpost1, Wlin1, blin1, hA,
                                               Nn, ntiles);
  // layer 2 (F=64)
  k_init_agg<<<iblocks, 256, 0, stream>>>(gS, gS2, gMn, gMx, gCnt, Nn);
  k_edge<64><<<eblocks, 128, 0, stream>>>(hA, eidx, eattr, Wpre2, ve0_2, ve1_2,
                                          gS, gS2, gMn, gMx, gCnt, Ee, etiles);
  k_node<64, true><<<nblocks, 64, 0, stream>>>(hA, gS, gS2, gMn, gMx, gCnt,
                                               Wpost2, bpost2, Wlin2, blin2,
                                               hB, Nn, ntiles);
  // layer 3 (F=64, no activation)
  k_init_agg<<<iblocks, 256, 0, stream>>>(gS, gS2, gMn, gMx, gCnt, Nn);
  k_edge<64><<<eblocks, 128, 0, stream>>>(hB, eidx, eattr, Wpre3, ve0_3, ve1_3,
                                          gS, gS2, gMn, gMx, gCnt, Ee, etiles);
  k_node<64, false><<<nblocks, 64, 0, stream>>>(hB, gS, gS2, gMn, gMx, gCnt,
                                                Wpost3, bpost3, Wlin3, blin3,
                                                hA, Nn, ntiles);
  // mean pool + head
  k_pool_zero<<<1, 1024, 0, stream>>>(gsum, gcnt);
  k_pool<<<(Nn + 1023) / 1024, 256, 0, stream>>>(hA, batch, gsum, gcnt, Nn);
  k_head<<<1, 1024, 0, stream>>>(gsum, gcnt, Wh1, bh1, Wh2, bh2, out);
}